// out_attention_44324062495124
// MI455X (gfx1250) — compile-verified
//
#include <hip/hip_runtime.h>

// ---------------------------------------------------------------------------
// out = softmax(10*tanh( (h_c W_Q) (H W_K)^T / sqrt(512) ))
// Rewritten via associativity:  qt[b] = (h_c[b] @ W_Q) @ W_K^T   (exact fp32)
//                               U[b,n] = dot(qt[b], H[b,n])
// Dominant cost: streaming H (1.07 GB) once -> HBM bound (~46us @ 23.3 TB/s).
// ---------------------------------------------------------------------------

typedef __attribute__((ext_vector_type(2))) float v2f;
typedef __attribute__((ext_vector_type(4))) float f4;
typedef __attribute__((ext_vector_type(8))) float v8f;

#define B_SZ   128
#define N_SZ   4096
#define DH     512
#define DK     512
#define DC     1536
#define NSPLIT 8      // N-splits per batch for the streaming kernel

// ---------------------------------------------------------------------------
// Generic fp32 WMMA GEMM: C[M,N] = A[M,K] * B[K,N], one wave per 16x16 tile.
// B element [k,n] is fetched as Bm[k*brs + n*bcs] so the same kernel handles
// W_Q (row-major) and W_K^T (brs=1, bcs=DK).
// Uses V_WMMA_F32_16X16X4_F32 (CDNA5 fp32 matrix pipe, exact fp32 math).
// A layout (16x4 f32): lane<16: M=lane, v0=K+0, v1=K+1 ; lane>=16: K+2/K+3.
// C/D layout (16x16 f32): VGPR r -> M = r (lanes 0-15), M = r+8 (lanes 16-31).
// ---------------------------------------------------------------------------
__global__ void __launch_bounds__(32)
wmma_gemm_f32_16x16x4(const float* __restrict__ A, int lda,
                      const float* __restrict__ Bm, int brs, int bcs,
                      float* __restrict__ C, int ldc, int Kdim) {
  const int lane = threadIdx.x & 31;
  const int hh   = lane >> 4;    // which half-wave (K pair select)
  const int l    = lane & 15;
  const int m0   = blockIdx.y * 16;
  const int n0   = blockIdx.x * 16;

  v8f acc = {};
  for (int k = 0; k < Kdim; k += 4) {
    const int ka = k + 2 * hh;
    v2f a, bv;
    a[0]  = A[(size_t)(m0 + l) * lda + ka + 0];
    a[1]  = A[(size_t)(m0 + l) * lda + ka + 1];
    bv[0] = Bm[(size_t)(ka + 0) * brs + (size_t)(n0 + l) * bcs];
    bv[1] = Bm[(size_t)(ka + 1) * brs + (size_t)(n0 + l) * bcs];
    acc = __builtin_amdgcn_wmma_f32_16x16x4_f32(
        /*neg_a=*/false, a, /*neg_b=*/false, bv,
        /*c_mod=*/(short)0, acc, /*reuse_a=*/false, /*reuse_b=*/false);
  }

#pragma unroll
  for (int r = 0; r < 8; ++r)
    C[(size_t)(m0 + r + 8 * hh) * ldc + (n0 + l)] = acc[r];
}

// ---------------------------------------------------------------------------
// Streaming kernel: one block per (N-split, batch). 512 threads = 16 wave32.
// Each wave cooperatively computes 32 row-dots: lane j holds elements
// [4j .. 4j+3] of each 128-element chunk -> 512B fully-coalesced b128 loads.
// H is read exactly once and is >> L2 (192MB) -> non-temporal loads.
// 10*tanh(x) in [-10,10] -> exp never overflows -> single-pass softmax (no max).
// ---------------------------------------------------------------------------
__global__ void __launch_bounds__(512)
scores_kernel(const float* __restrict__ H, const float* __restrict__ Qt,
              float* __restrict__ out_e, float* __restrict__ partial) {
  const int split = blockIdx.x;   // 0..NSPLIT-1
  const int b     = blockIdx.y;   // batch

  __shared__ __align__(16) float q[DH];
  __shared__ float wsum[16];

  const int tid = threadIdx.x;
  q[tid] = Qt[(size_t)b * DH + tid];
  __syncthreads();

  const int wave = tid >> 5;
  const int lane = tid & 31;
  const float* __restrict__ Hb = H + (size_t)b * N_SZ * DH;
  const float scale = 0.04419417382415922f;  // 1/sqrt(512)

  float psum = 0.0f;
  const int nbase = split * (N_SZ / NSPLIT) + wave * 32;

  for (int r = 0; r < 32; ++r) {
    const int n = nbase + r;
    const float* hrow = Hb + (size_t)n * DH;
    float acc = 0.0f;
#pragma unroll
    for (int c = 0; c < 4; ++c) {
      f4 hv = __builtin_nontemporal_load((const f4*)(hrow + c * 128 + lane * 4));
      f4 qv = *(const f4*)(&q[c * 128 + lane * 4]);
      acc = fmaf(hv[0], qv[0], acc);
      acc = fmaf(hv[1], qv[1], acc);
      acc = fmaf(hv[2], qv[2], acc);
      acc = fmaf(hv[3], qv[3], acc);
    }
    // wave32 tree reduction
#pragma unroll
    for (int off = 16; off > 0; off >>= 1)
      acc += __shfl_down(acc, off, 32);
    if (lane == 0) {
      float s = 10.0f * tanhf(acc * scale);
      float e = __expf(s);               // bounded by e^10, safe without max-sub
      out_e[(size_t)b * N_SZ + n] = e;
      psum += e;
    }
  }

  if (lane == 0) wsum[wave] = psum;
  __syncthreads();
  if (tid == 0) {
    float t = 0.0f;
#pragma unroll
    for (int w = 0; w < 16; ++w) t += wsum[w];
    partial[b * NSPLIT + split] = t;
  }
}

// ---------------------------------------------------------------------------
// Final normalization: reduce the NSPLIT partial sums, scale the output row.
// ---------------------------------------------------------------------------
__global__ void __launch_bounds__(256)
normalize_kernel(const float* __restrict__ partial, float* __restrict__ out) {
  const int b = blockIdx.x;
  __shared__ float inv_total;
  if (threadIdx.x == 0) {
    float t = 0.0f;
#pragma unroll
    for (int i = 0; i < NSPLIT; ++i) t += partial[b * NSPLIT + i];
    inv_total = 1.0f / t;
  }
  __syncthreads();
  const float inv = inv_total;
  for (int i = threadIdx.x; i < N_SZ; i += 256)
    out[(size_t)b * N_SZ + i] *= inv;
}

// ---------------------------------------------------------------------------
extern "C" void kernel_launch(void* const* d_in, const int* in_sizes, int n_in,
                              void* d_out, int out_size, void* d_ws, size_t ws_size,
                              hipStream_t stream) {
  const float* h_c = (const float*)d_in[0];   // [128, 1, 1536]
  const float* H   = (const float*)d_in[1];   // [128, 4096, 512]
  const float* W_Q = (const float*)d_in[2];   // [1536, 512]
  const float* W_K = (const float*)d_in[3];   // [512, 512]
  float* out = (float*)d_out;                 // [128, 4096]

  float* Qall    = (float*)d_ws;              // [128, 512]
  float* Qt      = Qall + B_SZ * DK;          // [128, 512]
  float* partial = Qt + B_SZ * DH;            // [128, NSPLIT]

  // Stage 1: Qall = h_c @ W_Q           (A:[128,1536], B row-major)
  dim3 g1(DK / 16, B_SZ / 16);
  wmma_gemm_f32_16x16x4<<<g1, 32, 0, stream>>>(h_c, DC, W_Q, DK, 1, Qall, DK, DC);

  // Stage 2: Qt = Qall @ W_K^T          (B[k,d] = W_K[d,k] -> brs=1, bcs=DK)
  dim3 g2(DH / 16, B_SZ / 16);
  wmma_gemm_f32_16x16x4<<<g2, 32, 0, stream>>>(Qall, DK, W_K, 1, DK, Qt, DH, DK);

  // Stage 3: stream H once, compute exp(10*tanh(U/sqrt(dk))) + partial sums
  dim3 g3(NSPLIT, B_SZ);
  scores_kernel<<<g3, 512, 0, stream>>>(H, Qt, out, partial);

  // Stage 4: normalize per batch row
  normalize_kernel<<<dim3(B_SZ), 256, 0, stream>>>(partial, out);
}